// Reranker_15530601742775
// MI455X (gfx1250) — compile-verified
//
#include <hip/hip_runtime.h>
#include <hip/hip_bf16.h>
#include <math.h>

typedef __attribute__((ext_vector_type(2))) float v2f;
typedef __attribute__((ext_vector_type(8))) float v8f;

#define NPASS 64
#define WAVES_PER_BLOCK 8
#define BLOCK_THREADS 256

// Stable softplus(-pd) on the native base-2 transcendental units:
//   softplus(x) = max(x,0) + ln2 * log2(1 + exp2(-|x| * log2e))
// ~6 VALU ops (2 TRANS) vs ~45 for expf+log1pf libcall expansion.
__device__ __forceinline__ float softplus_neg(float pd) {
  const float x = -pd;
  const float LOG2E = 1.4426950408889634f;
  const float LN2   = 0.6931471805599453f;
#if __has_builtin(__builtin_amdgcn_exp2f) && __has_builtin(__builtin_amdgcn_logf)
  const float e = __builtin_amdgcn_exp2f(-fabsf(x) * LOG2E);  // v_exp_f32
  const float l = __builtin_amdgcn_logf(1.0f + e);            // v_log_f32 (log2)
  return fmaxf(x, 0.0f) + LN2 * l;
#else
  return fmaxf(x, 0.0f) + __logf(1.0f + __expf(-fabsf(x)));
#endif
}

// Stage 1: one wave32 per row. Stage row in LDS, compute tie-aware rank y_true,
// then compute all 64x64 pair tiles via V_WMMA_F32_16X16X4_F32 outer-difference
// trick: tile(pred_diff) = A x B with A=[p_i, -1, 0, 0] (16x4), B=[1; p_j; 0; 0] (4x16).
__global__ __launch_bounds__(BLOCK_THREADS)
void ranknet_partial_kernel(const float* __restrict__ logits,
                            const int* __restrict__ kd_labels,
                            float* __restrict__ partial,
                            int B, int nblocks) {
  __shared__ float p_s[WAVES_PER_BLOCK][NPASS];
  __shared__ float t_s[WAVES_PER_BLOCK][NPASS];
  __shared__ int   l_s[WAVES_PER_BLOCK][NPASS];
  __shared__ float wsum[WAVES_PER_BLOCK];
  __shared__ float wcnt[WAVES_PER_BLOCK];

  const int tid  = threadIdx.x;
  const int wave = tid >> 5;
  const int lane = tid & 31;
  const int row  = blockIdx.x * WAVES_PER_BLOCK + wave;
  const bool valid = row < B;   // wave-uniform

  if (valid) {
    const float* pr = logits    + (size_t)row * NPASS;
    const int*   lr = kd_labels + (size_t)row * NPASS;
    p_s[wave][lane]      = pr[lane];
    p_s[wave][lane + 32] = pr[lane + 32];
    l_s[wave][lane]      = lr[lane];
    l_s[wave][lane + 32] = lr[lane + 32];
  }
  __syncthreads();

  // Tie-aware rank: y_true[j] = 1/(start_of_run(j)+1), run = adjacent equal labels.
  if (valid) {
#pragma unroll
    for (int e = 0; e < 2; ++e) {
      const int j   = lane + e * 32;
      const int lab = l_s[wave][j];
      int k = j;
      while (k > 0 && l_s[wave][k - 1] == lab) --k;
      t_s[wave][j] = 1.0f / (float)(k + 1);
    }
  }
  __syncthreads();

  float s   = 0.0f;
  int   cnt = 0;
  if (valid) {
    const int  lid = lane & 15;
    const bool lo  = (lane < 16);   // lanes 0-15 carry K=0,1 rows; upper half = zero rows
    for (int tj = 0; tj < 4; ++tj) {
      const float pj  = p_s[wave][tj * 16 + lid];
      const float tjv = t_s[wave][tj * 16 + lid];
      v2f bp, bt;
      bp[0] = lo ? 1.0f : 0.0f;  bp[1] = lo ? pj  : 0.0f;   // B rows: [1 ; p_j]
      bt[0] = lo ? 1.0f : 0.0f;  bt[1] = lo ? tjv : 0.0f;   // B rows: [1 ; t_j]
#pragma unroll
      for (int ti = 0; ti < 4; ++ti) {
        const float pi  = p_s[wave][ti * 16 + lid];
        const float tiv = t_s[wave][ti * 16 + lid];
        v2f ap, at;
        ap[0] = lo ? pi  : 0.0f;  ap[1] = lo ? -1.0f : 0.0f; // A cols: [p_i, -1]
        at[0] = lo ? tiv : 0.0f;  at[1] = lo ? -1.0f : 0.0f; // A cols: [t_i, -1]
        v8f cz = {0.f, 0.f, 0.f, 0.f, 0.f, 0.f, 0.f, 0.f};
        // D[m,n] = p[ti*16+m] - p[tj*16+n]  (pure f32 WMMA, no precision loss)
        v8f pd = __builtin_amdgcn_wmma_f32_16x16x4_f32(
            false, ap, false, bp, (short)0, cz, false, false);
        v8f td = __builtin_amdgcn_wmma_f32_16x16x4_f32(
            false, at, false, bt, (short)0, cz, false, false);
#pragma unroll
        for (int v = 0; v < 8; ++v) {
          const float sp = softplus_neg(pd[v]);
          const bool  m  = td[v] > 0.0f;   // y_true finite -> isinf filter is vacuous
          s   = m ? (s + sp) : s;
          cnt = m ? (cnt + 1) : cnt;
        }
      }
    }
  }

  // In-wave deterministic butterfly reduction (wave32).
  for (int off = 16; off >= 1; off >>= 1) {
    s   += __shfl_xor(s, off, 32);
    cnt += __shfl_xor(cnt, off, 32);
  }
  if (lane == 0) { wsum[wave] = s; wcnt[wave] = (float)cnt; }
  __syncthreads();
  if (tid == 0) {
    float S = 0.f, C = 0.f;
#pragma unroll
    for (int w = 0; w < WAVES_PER_BLOCK; ++w) { S += wsum[w]; C += wcnt[w]; }
    partial[blockIdx.x]           = S;
    partial[nblocks + blockIdx.x] = C;  // counts exact in f32 (< 2^24)
  }
}

// Stage 2: single block, fixed-order tree reduction -> loss = sum / count.
__global__ __launch_bounds__(256)
void ranknet_finalize_kernel(const float* __restrict__ partial, int nblocks,
                             float* __restrict__ out) {
  __shared__ float ss[256];
  __shared__ float cc[256];
  const int t = threadIdx.x;
  float s = 0.f, c = 0.f;
  for (int i = t; i < nblocks; i += 256) {
    s += partial[i];
    c += partial[nblocks + i];
  }
  ss[t] = s; cc[t] = c;
  __syncthreads();
  for (int off = 128; off > 0; off >>= 1) {
    if (t < off) { ss[t] += ss[t + off]; cc[t] += cc[t + off]; }
    __syncthreads();
  }
  if (t == 0) out[0] = ss[0] / cc[0];
}

extern "C" void kernel_launch(void* const* d_in, const int* in_sizes, int n_in,
                              void* d_out, int out_size, void* d_ws, size_t ws_size,
                              hipStream_t stream) {
  const float* logits = (const float*)d_in[0];   // [B*n, 1] f32
  const int*   kd     = (const int*)d_in[1];     // [B, n] i32
  (void)n_in; (void)out_size; (void)ws_size;

  const int total   = in_sizes[0];
  const int B       = total / NPASS;             // n = 64 (train_n_passages)
  const int nblocks = (B + WAVES_PER_BLOCK - 1) / WAVES_PER_BLOCK;

  float* partial = (float*)d_ws;                 // 2 * nblocks floats of scratch

  ranknet_partial_kernel<<<nblocks, BLOCK_THREADS, 0, stream>>>(
      logits, kd, partial, B, nblocks);
  ranknet_finalize_kernel<<<1, 256, 0, stream>>>(partial, nblocks, (float*)d_out);
}